// InhibitionLayer_386547057370
// MI455X (gfx1250) — compile-verified
//
#include <hip/hip_runtime.h>
#include <stdint.h>

// ---------------------------------------------------------------------------
// InhibitionLayer: leaky integrate + global top-128 WTA + refractory update.
// N = 2^25 fp32 neurons. Memory-bound: min traffic 768 MB -> ~33us @ 23.3TB/s;
// selection passes add ~512MB -> ~55us total, still HBM-bound.
// Pipeline: hist(top12 of order-preserving u32) -> scan -> gather candidates
// -> exact radix+tie select -> fused elementwise finalize.
// CDNA5 paths: double-buffered global_load_async_to_lds_b128 + s_wait_asynccnt
// pipelining, ds_add_u32 LDS histograms, global_prefetch_b8, NT cache hints.
// ---------------------------------------------------------------------------

#define KTOP 128
#define HIST1_BINS 4096
#define CAP 65536  // candidate list capacity (plenty for Gaussian tails)

typedef __attribute__((ext_vector_type(4))) float f4;  // native vector: NT-able

// ws layout (u32 units):
//   [0] b1   [1] c_above   [2] cand_count   [3] t   [4] cutoff   [5..7] pad
//   [8 .. 8+4096)        level-1 histogram
//   [8192 .. 8192+2*CAP) candidate (u, idx) pairs
// total ws use: (8192 + 2*65536)*4 = 544 KB

__device__ __forceinline__ float lif(float x, float v) {
  return v + (x - v) / 2.0f;  // TAU = 2: exact halving, matches reference bits
}

// order-preserving float->u32 map (larger float => larger u32)
__device__ __forceinline__ unsigned f2u(float f) {
  unsigned b = __float_as_uint(f);
  return b ^ (unsigned)(((int)b >> 31) | 0x80000000);
}

__global__ void k_init(unsigned* __restrict__ ws) {
  int i = blockIdx.x * blockDim.x + threadIdx.x;
  if (i < 8 + HIST1_BINS) ws[i] = 0u;
}

// Pass 1: histogram of top-12 bits of u.
// Double-buffered async global->LDS staging: issue tile i+1, wait asynccnt<=2
// (in-order completion guarantees tile i landed), consume tile i.
__global__ void __launch_bounds__(256) k_hist1(const float* __restrict__ x,
                                               const float* __restrict__ v,
                                               unsigned* __restrict__ ws, int n) {
  __shared__ unsigned lh[HIST1_BINS];
  __shared__ f4 sx[2][256];
  __shared__ f4 sv[2][256];
  unsigned* hist = ws + 8;
  for (int i = threadIdx.x; i < HIST1_BINS; i += blockDim.x) lh[i] = 0u;
  __syncthreads();

  const int vecN = n >> 2;
  const f4* x4 = (const f4*)x;
  const f4* v4 = (const f4*)v;
  // Per-lane LDS byte offsets (flat LDS addresses keep the offset in [31:0]).
  const unsigned lx0 = (unsigned)(size_t)&sx[0][threadIdx.x];
  const unsigned lx1 = (unsigned)(size_t)&sx[1][threadIdx.x];
  const unsigned lv0 = (unsigned)(size_t)&sv[0][threadIdx.x];
  const unsigned lv1 = (unsigned)(size_t)&sv[1][threadIdx.x];

  const int stride = gridDim.x * blockDim.x;
  int vid = blockIdx.x * blockDim.x + threadIdx.x;
  int buf = 0;
  if (vid < vecN) {  // prologue: fill buffer 0
    asm volatile("global_load_async_to_lds_b128 %0, %1, off"
                 :: "v"(lx0), "v"(x4 + vid) : "memory");
    asm volatile("global_load_async_to_lds_b128 %0, %1, off"
                 :: "v"(lv0), "v"(v4 + vid) : "memory");
  }
  for (; vid < vecN; vid += stride) {
    const int nxt = vid + stride;
    if (nxt < vecN) {
      const unsigned nlx = buf ? lx0 : lx1;
      const unsigned nlv = buf ? lv0 : lv1;
      asm volatile("global_load_async_to_lds_b128 %0, %1, off"
                   :: "v"(nlx), "v"(x4 + nxt) : "memory");
      asm volatile("global_load_async_to_lds_b128 %0, %1, off"
                   :: "v"(nlv), "v"(v4 + nxt) : "memory");
      asm volatile("s_wait_asynccnt 2" ::: "memory");  // older pair done
    } else {
      asm volatile("s_wait_asynccnt 0" ::: "memory");  // drain
    }
    f4 xx = sx[buf][threadIdx.x];
    f4 vv = sv[buf][threadIdx.x];
    atomicAdd(&lh[f2u(lif(xx.x, vv.x)) >> 20], 1u);
    atomicAdd(&lh[f2u(lif(xx.y, vv.y)) >> 20], 1u);
    atomicAdd(&lh[f2u(lif(xx.z, vv.z)) >> 20], 1u);
    atomicAdd(&lh[f2u(lif(xx.w, vv.w)) >> 20], 1u);
    buf ^= 1;
  }
  // scalar tail (n % 4)
  for (int i = (vecN << 2) + blockIdx.x * blockDim.x + threadIdx.x; i < n; i += stride)
    atomicAdd(&hist[f2u(lif(x[i], v[i])) >> 20], 1u);

  __syncthreads();
  for (int i = threadIdx.x; i < HIST1_BINS; i += blockDim.x)
    if (lh[i]) atomicAdd(&hist[i], lh[i]);
}

// Find the bin holding the K-th largest value; count strictly above it.
__global__ void k_scan1(unsigned* __restrict__ ws) {
  if (threadIdx.x == 0) {
    const unsigned* hist = ws + 8;
    unsigned suffix = 0, b1 = 0;
    for (int i = HIST1_BINS - 1; i >= 0; --i) {
      unsigned c = hist[i];
      if (suffix + c >= (unsigned)KTOP) { b1 = (unsigned)i; break; }
      suffix += c;
    }
    ws[0] = b1;      // threshold bin
    ws[1] = suffix;  // c_above (< K)
    ws[2] = 0u;      // candidate counter
  }
}

// Collect all elements in bins >= b1 into the candidate list.
__global__ void __launch_bounds__(256) k_gather(const float* __restrict__ x,
                                                const float* __restrict__ v,
                                                unsigned* __restrict__ ws, int n) {
  const unsigned b1 = ws[0];
  unsigned* cands = ws + 8192;
  const int stride = gridDim.x * blockDim.x;
  for (int i = blockIdx.x * blockDim.x + threadIdx.x; i < n; i += stride) {
    __builtin_prefetch(x + i + stride, 0, 0);
    __builtin_prefetch(v + i + stride, 0, 0);
    unsigned u = f2u(lif(x[i], v[i]));
    if ((u >> 20) >= b1) {
      unsigned pos = atomicAdd(&ws[2], 1u);
      if (pos < (unsigned)CAP) {
        cands[2 * pos]     = u;
        cands[2 * pos + 1] = (unsigned)i;
      }
    }
  }
}

// Exact threshold t (K-th largest u) + lowest-index tie cutoff. One block.
__global__ void __launch_bounds__(1024) k_select(unsigned* __restrict__ ws, int n) {
  __shared__ unsigned h[1024];
  __shared__ unsigned s_sel, s_pfx, s_t, s_kt, s_lo, s_hi, s_cnt;
  const unsigned* cands = ws + 8192;
  unsigned M = ws[2]; if (M > (unsigned)CAP) M = (unsigned)CAP;
  const unsigned b1 = ws[0];
  const unsigned k2 = (unsigned)KTOP - ws[1];  // rank needed inside bin b1 (>=1)

  // stage 1: radix on bits 19:10 within bin b1
  h[threadIdx.x] = 0u;
  __syncthreads();
  for (unsigned j = threadIdx.x; j < M; j += 1024u) {
    unsigned u = cands[2 * j];
    if ((u >> 20) == b1) atomicAdd(&h[(u >> 10) & 1023u], 1u);
  }
  __syncthreads();
  if (threadIdx.x == 0) {
    unsigned suffix = 0, sel = 0;
    for (int i = 1023; i >= 0; --i) {
      unsigned c = h[i];
      if (suffix + c >= k2) { sel = (unsigned)i; break; }
      suffix += c;
    }
    s_sel = sel; s_pfx = suffix;
  }
  __syncthreads();
  const unsigned s1 = s_sel;
  const unsigned k3 = k2 - s_pfx;
  __syncthreads();

  // stage 2: radix on bits 9:0 within the 22-bit group
  h[threadIdx.x] = 0u;
  __syncthreads();
  const unsigned top22 = (b1 << 10) | s1;
  for (unsigned j = threadIdx.x; j < M; j += 1024u) {
    unsigned u = cands[2 * j];
    if ((u >> 10) == top22) atomicAdd(&h[u & 1023u], 1u);
  }
  __syncthreads();
  if (threadIdx.x == 0) {
    unsigned suffix = 0, sel = 0;
    for (int i = 1023; i >= 0; --i) {
      unsigned c = h[i];
      if (suffix + c >= k3) { sel = (unsigned)i; break; }
      suffix += c;
    }
    s_t  = (top22 << 10) | sel;  // exact K-th largest as ordered u32
    s_kt = k3 - suffix;          // how many ties win (>=1)
    s_lo = 0u; s_hi = (unsigned)(n - 1);
  }
  __syncthreads();
  const unsigned t = s_t, kt = s_kt;

  // stage 3: binary search smallest C with #{u==t && idx<=C} >= kt
  while (true) {
    unsigned lo = s_lo, hi = s_hi;
    if (lo >= hi) break;
    unsigned mid = lo + ((hi - lo) >> 1);
    if (threadIdx.x == 0) s_cnt = 0u;
    __syncthreads();
    unsigned c = 0;
    for (unsigned j = threadIdx.x; j < M; j += 1024u)
      if (cands[2 * j] == t && cands[2 * j + 1] <= mid) c++;
    if (c) atomicAdd(&s_cnt, c);
    __syncthreads();
    if (threadIdx.x == 0) { if (s_cnt >= kt) s_hi = mid; else s_lo = mid + 1u; }
    __syncthreads();
  }
  if (threadIdx.x == 0) { ws[3] = t; ws[4] = s_hi; }
}

struct NeuronOut { float y, vo, ro; };

__device__ __forceinline__ NeuronOut neuron(float xx, float vv, float rr,
                                            unsigned idx, unsigned t,
                                            unsigned cutoff) {
  float vn = lif(xx, vv);
  unsigned u = f2u(vn);
  bool winner = (u > t) || (u == t && idx <= cutoff);
  bool fire = winner && (vn > 1.0f) && (rr == 0.0f);
  NeuronOut o;
  o.y = fire ? 1.0f : 0.0f;
  float ra = fire ? 5.0f : rr;                 // REFRACTORY
  o.vo = (fire || ra > 0.0f) ? 0.0f : vn;
  o.ro = (ra > 0.0f) ? (ra - 1.0f) : 0.0f;
  return o;
}

// Fused finalize: winner mask + fire + refractory. Outputs are write-once
// streams: nontemporal stores keep them from displacing x/v in L2.
__global__ void __launch_bounds__(256) k_final(const float* __restrict__ x,
                                               const float* __restrict__ v,
                                               const float* __restrict__ r,
                                               float* __restrict__ out,
                                               const unsigned* __restrict__ ws, int n) {
  const unsigned t = ws[3];
  const unsigned cutoff = ws[4];
  float* y  = out;
  float* vo = out + (size_t)n;
  float* ro = out + 2 * (size_t)n;
  const int vecN = n >> 2;
  const f4* x4 = (const f4*)x;
  const f4* v4 = (const f4*)v;
  const f4* r4 = (const f4*)r;
  f4* y4  = (f4*)y;
  f4* vo4 = (f4*)vo;
  f4* ro4 = (f4*)ro;
  const int stride = gridDim.x * blockDim.x;
  for (int vid = blockIdx.x * blockDim.x + threadIdx.x; vid < vecN; vid += stride) {
    __builtin_prefetch(x4 + vid + stride, 0, 0);
    __builtin_prefetch(v4 + vid + stride, 0, 0);
    f4 xx = x4[vid];
    f4 vv = v4[vid];
    f4 rr = __builtin_nontemporal_load(r4 + vid);  // read-once stream
    unsigned base = (unsigned)vid << 2;
    NeuronOut o0 = neuron(xx.x, vv.x, rr.x, base + 0u, t, cutoff);
    NeuronOut o1 = neuron(xx.y, vv.y, rr.y, base + 1u, t, cutoff);
    NeuronOut o2 = neuron(xx.z, vv.z, rr.z, base + 2u, t, cutoff);
    NeuronOut o3 = neuron(xx.w, vv.w, rr.w, base + 3u, t, cutoff);
    f4 oy;  oy.x  = o0.y;  oy.y  = o1.y;  oy.z  = o2.y;  oy.w  = o3.y;
    f4 ov;  ov.x  = o0.vo; ov.y  = o1.vo; ov.z  = o2.vo; ov.w  = o3.vo;
    f4 orr; orr.x = o0.ro; orr.y = o1.ro; orr.z = o2.ro; orr.w = o3.ro;
    __builtin_nontemporal_store(oy,  y4 + vid);
    __builtin_nontemporal_store(ov,  vo4 + vid);
    __builtin_nontemporal_store(orr, ro4 + vid);
  }
  for (int i = (vecN << 2) + blockIdx.x * blockDim.x + threadIdx.x; i < n; i += stride) {
    NeuronOut o = neuron(x[i], v[i], r[i], (unsigned)i, t, cutoff);
    y[i] = o.y; vo[i] = o.vo; ro[i] = o.ro;
  }
}

extern "C" void kernel_launch(void* const* d_in, const int* in_sizes, int n_in,
                              void* d_out, int out_size, void* d_ws, size_t ws_size,
                              hipStream_t stream) {
  const float* x = (const float*)d_in[0];
  const float* v = (const float*)d_in[1];
  const float* r = (const float*)d_in[2];
  float* out = (float*)d_out;
  unsigned* ws = (unsigned*)d_ws;
  const int n = in_sizes[0];
  const int vecN = n >> 2;

  int initBlocks = (8 + HIST1_BINS + 255) / 256;
  k_init<<<initBlocks, 256, 0, stream>>>(ws);

  int blocks = (vecN + 255) / 256;
  if (blocks > 8192) blocks = 8192;
  if (blocks < 1) blocks = 1;
  k_hist1<<<blocks, 256, 0, stream>>>(x, v, ws, n);

  k_scan1<<<1, 64, 0, stream>>>(ws);

  int gblocks = (n + 255) / 256;
  if (gblocks > 8192) gblocks = 8192;
  if (gblocks < 1) gblocks = 1;
  k_gather<<<gblocks, 256, 0, stream>>>(x, v, ws, n);

  k_select<<<1, 1024, 0, stream>>>(ws, n);

  k_final<<<blocks, 256, 0, stream>>>(x, v, r, out, ws, n);
}